// MLACompatibleAttention_37271726194773
// MI455X (gfx1250) — compile-verified
//
#include <hip/hip_runtime.h>
#include <math.h>

typedef __attribute__((ext_vector_type(2))) float v2f;
typedef __attribute__((ext_vector_type(8))) float v8f;

#define B_     8
#define D_     4096
#define H_     32
#define HD_    128
#define KV_    512
#define PAST_  8191
#define NCH_   8
#define SC_    1024          // positions per chunk
#define TILE_  32            // positions per inner tile
#define ITERS_ (SC_ / TILE_)
#define SCALE_ 0.08838834764831845f   // 1/sqrt(128)

// ---------------------------------------------------------------- zero init
__global__ void mla_zero_kernel(float* __restrict__ p, int n) {
    int i = blockIdx.x * blockDim.x + threadIdx.x;
    if (i < n) p[i] = 0.0f;
}

// -------------------------------------------------- batched GEMV projections
// out[b][h][0:KV] += sum_{d in chunk} W[h][d][k] * x[b][d]
// grid = (nmat, 4 d-chunks of 1024), block = 128 (one float4 of k per thread)
__global__ void mla_proj8_kernel(const float* __restrict__ W, size_t mat_stride,
                                 const float* __restrict__ x,
                                 float* __restrict__ out, int out_bstride) {
    __shared__ float x_s[B_ * 1024];
    const int h  = blockIdx.x;
    const int d0 = blockIdx.y * 1024;
    for (int i = threadIdx.x; i < B_ * 1024; i += blockDim.x) {
        int b = i >> 10, d = i & 1023;
        x_s[i] = x[b * D_ + d0 + d];
    }
    __syncthreads();

    const float* Wm = W + (size_t)h * mat_stride;
    const int k4 = threadIdx.x;          // float4 column index, covers KV=512
    float acc[B_][4] = {};
    for (int dd = 0; dd < 1024; ++dd) {
        const float4 w = ((const float4*)(Wm + (size_t)(d0 + dd) * KV_))[k4];
#pragma unroll
        for (int b = 0; b < B_; ++b) {
            const float xv = x_s[b * 1024 + dd];
            acc[b][0] = fmaf(w.x, xv, acc[b][0]);
            acc[b][1] = fmaf(w.y, xv, acc[b][1]);
            acc[b][2] = fmaf(w.z, xv, acc[b][2]);
            acc[b][3] = fmaf(w.w, xv, acc[b][3]);
        }
    }
#pragma unroll
    for (int b = 0; b < B_; ++b)
#pragma unroll
        for (int c = 0; c < 4; ++c)
            atomicAdd(&out[(size_t)b * out_bstride + h * KV_ + k4 * 4 + c], acc[b][c]);
}

// ------------------------------------------------------ flash decode (WMMA)
// grid = (B, NCH), block = 128 (4 waves). wave = (mi = head-tile, hs = n/k half)
__global__ __launch_bounds__(128, 1)
void mla_flash_kernel(const float* __restrict__ past_c, const float* __restrict__ c_new,
                      const float* __restrict__ q_abs,
                      float* __restrict__ ctx_part, float* __restrict__ ml_part) {
    extern __shared__ float smem[];
    float* q_s   = smem;                    // 32*512
    float* c_s   = q_s + H_ * KV_;          // 32*512
    float* p_s   = c_s + TILE_ * KV_;       // 32*32
    float* m_s   = p_s + H_ * TILE_;        // 32
    float* l_s   = m_s + H_;                // 32
    float* mn_s  = l_s + H_;                // 32
    float* sc_s  = mn_s + H_;               // 32
    float* mpart = sc_s + H_;               // 2*32
    float* lpart = mpart + 2 * H_;          // 2*32

    const int b    = blockIdx.x;
    const int ch   = blockIdx.y;
    const int tid  = threadIdx.x;
    const int lane = tid & 31;
    const int wave = tid >> 5;
    const int mi   = wave >> 1;             // which 16-head tile
    const int hs   = wave & 1;              // scores: n-tile; ctx: k-half
    const int l15  = lane & 15;
    const int lh   = lane >> 4;             // lane half (0/1)
    const int koff = 2 * lh;                // A-operand K offset for this half

    // stage q_abs[b] (32x512) into LDS
    const float4* qsrc = (const float4*)(q_abs + (size_t)b * H_ * KV_);
    float4* qdst = (float4*)q_s;
    for (int i = tid; i < H_ * KV_ / 4; i += 128) qdst[i] = qsrc[i];
    if (tid < H_) { m_s[tid] = -INFINITY; l_s[tid] = 0.0f; }

    v8f ctx[16];
#pragma unroll
    for (int n = 0; n < 16; ++n)
#pragma unroll
        for (int r = 0; r < 8; ++r) ctx[n][r] = 0.0f;

    const int hbase = mi * 16;
    const int nbase = hs * 16;
    const int kbase = hs * 256;
    const size_t pc_base = (size_t)b * PAST_ * KV_;

    for (int it = 0; it < ITERS_; ++it) {
        __syncthreads();   // protect c_s / p_s against previous iteration readers
        const int s0 = ch * SC_ + it * TILE_;

        // ---- stream c tile (32 x 512) into LDS, coalesced b128 ----
        for (int f = tid; f < TILE_ * KV_ / 4; f += 128) {
            const int row = f >> 7;       // /128 float4 per row
            const int c4  = f & 127;
            const int s   = s0 + row;
            float4 v;
            if (s < PAST_) v = ((const float4*)(past_c + pc_base + (size_t)s * KV_))[c4];
            else           v = ((const float4*)(c_new + (size_t)b * KV_))[c4];
            ((float4*)c_s)[f] = v;
            if (s + TILE_ < PAST_)
                __builtin_prefetch(past_c + pc_base + (size_t)(s + TILE_) * KV_ + c4 * 4, 0, 1);
        }
        __syncthreads();

        // ---- scores: C[16h x 16pos] = q[16h x 512] * c^T[512 x 16pos] ----
        v8f sacc = {0.f, 0.f, 0.f, 0.f, 0.f, 0.f, 0.f, 0.f};
        const int arow = (hbase + l15) * KV_;
        const int crow = (nbase + l15) * KV_;
#pragma unroll 4
        for (int kk = 0; kk < KV_ / 4; ++kk) {
            const int k = kk * 4;
            v2f a;  a.x  = q_s[arow + k + koff];  a.y  = q_s[arow + k + koff + 1];
            v2f bb; bb.x = c_s[crow + k + lh];    bb.y = c_s[crow + k + lh + 2];
            sacc = __builtin_amdgcn_wmma_f32_16x16x4_f32(false, a, false, bb,
                                                         (short)0, sacc, false, false);
        }

        // ---- online softmax: row max (16 lanes within a half share a row) ----
        float tmax[8];
#pragma unroll
        for (int r = 0; r < 8; ++r) {
            float v = sacc[r] * SCALE_;
            sacc[r] = v;
            v = fmaxf(v, __shfl_xor(v, 1, 32));
            v = fmaxf(v, __shfl_xor(v, 2, 32));
            v = fmaxf(v, __shfl_xor(v, 4, 32));
            v = fmaxf(v, __shfl_xor(v, 8, 32));
            tmax[r] = v;
        }
        if (l15 == 0) {
#pragma unroll
            for (int r = 0; r < 8; ++r) mpart[hs * H_ + hbase + r + 8 * lh] = tmax[r];
        }
        __syncthreads();
        if (tid < H_) {
            const float mo = m_s[tid];
            const float mn = fmaxf(mo, fmaxf(mpart[tid], mpart[H_ + tid]));
            mn_s[tid] = mn;
            const float scl = __expf(mo - mn);
            sc_s[tid] = scl;
            m_s[tid]  = mn;
            l_s[tid] *= scl;
        }
        __syncthreads();

        // ---- p = exp(s - m), row sums, stage p into LDS in (h,pos) layout ----
        float tsum[8];
#pragma unroll
        for (int r = 0; r < 8; ++r) {
            const int h = hbase + r + 8 * lh;
            float p = __expf(sacc[r] - mn_s[h]);
            p_s[h * TILE_ + nbase + l15] = p;
            p += __shfl_xor(p, 1, 32);
            p += __shfl_xor(p, 2, 32);
            p += __shfl_xor(p, 4, 32);
            p += __shfl_xor(p, 8, 32);
            tsum[r] = p;
        }
        if (l15 == 0) {
#pragma unroll
            for (int r = 0; r < 8; ++r) lpart[hs * H_ + hbase + r + 8 * lh] = tsum[r];
        }
        __syncthreads();
        if (tid < H_) l_s[tid] += lpart[tid] + lpart[H_ + tid];

        // ---- ctx: rescale then C[16h x 256k] += p[16h x 32pos] * c[32pos x 256k] ----
        float srg[8];
#pragma unroll
        for (int r = 0; r < 8; ++r) srg[r] = sc_s[hbase + r + 8 * lh];
#pragma unroll
        for (int n = 0; n < 16; ++n)
#pragma unroll
            for (int r = 0; r < 8; ++r) ctx[n][r] *= srg[r];

        const int prow = (hbase + l15) * TILE_;
#pragma unroll
        for (int kk = 0; kk < TILE_ / 4; ++kk) {
            const int kp = kk * 4;
            v2f a; a.x = p_s[prow + kp + koff]; a.y = p_s[prow + kp + koff + 1];
            const int crow0 = (kp + lh) * KV_;
#pragma unroll
            for (int n = 0; n < 16; ++n) {
                const int col = kbase + n * 16 + l15;
                v2f bb; bb.x = c_s[crow0 + col]; bb.y = c_s[crow0 + 2 * KV_ + col];
                ctx[n] = __builtin_amdgcn_wmma_f32_16x16x4_f32(false, a, false, bb,
                                                               (short)0, ctx[n], false, false);
            }
        }
    }

    __syncthreads();
    // ---- write chunk partials ----
    float* cp = ctx_part + ((size_t)b * NCH_ + ch) * H_ * KV_;
#pragma unroll
    for (int n = 0; n < 16; ++n) {
        const int col = kbase + n * 16 + l15;
#pragma unroll
        for (int r = 0; r < 8; ++r) {
            const int h = hbase + r + 8 * lh;
            cp[(size_t)h * KV_ + col] = ctx[n][r];
        }
    }
    if (tid < H_) {
        ml_part[((b * NCH_ + ch) * 2 + 0) * H_ + tid] = m_s[tid];
        ml_part[((b * NCH_ + ch) * 2 + 1) * H_ + tid] = l_s[tid];
    }
}

// ------------------------------------------------------- combine chunk parts
__global__ void mla_combine_kernel(const float* __restrict__ ctx_part,
                                   const float* __restrict__ ml_part,
                                   float* __restrict__ ctx_comb) {
    __shared__ float wts[NCH_];
    const int b = blockIdx.x, h = blockIdx.y;
    if (threadIdx.x == 0) {
        float M = -INFINITY;
        for (int c = 0; c < NCH_; ++c)
            M = fmaxf(M, ml_part[((b * NCH_ + c) * 2 + 0) * H_ + h]);
        float L = 0.0f;
        for (int c = 0; c < NCH_; ++c) {
            const float mc = ml_part[((b * NCH_ + c) * 2 + 0) * H_ + h];
            const float lc = ml_part[((b * NCH_ + c) * 2 + 1) * H_ + h];
            L += lc * __expf(mc - M);
        }
        const float invL = 1.0f / L;
        for (int c = 0; c < NCH_; ++c) {
            const float mc = ml_part[((b * NCH_ + c) * 2 + 0) * H_ + h];
            wts[c] = __expf(mc - M) * invL;
        }
    }
    __syncthreads();
    for (int k = threadIdx.x; k < KV_; k += blockDim.x) {
        float acc = 0.0f;
        for (int c = 0; c < NCH_; ++c)
            acc = fmaf(ctx_part[(((size_t)b * NCH_ + c) * H_ + h) * KV_ + k], wts[c], acc);
        ctx_comb[((size_t)b * H_ + h) * KV_ + k] = acc;
    }
}

// -------------------------------------------- V up-projection (absorbed)
__global__ void mla_vup_kernel(const float* __restrict__ ctx_comb,
                               const float* __restrict__ w_v,
                               float* __restrict__ oh) {
    __shared__ float c_s[KV_];
    const int b = blockIdx.x, h = blockIdx.y;
    for (int i = threadIdx.x; i < KV_; i += blockDim.x)
        c_s[i] = ctx_comb[((size_t)b * H_ + h) * KV_ + i];
    __syncthreads();
    const int j = threadIdx.x;    // 0..127
    float acc = 0.0f;
    for (int k = 0; k < KV_; ++k)
        acc = fmaf(c_s[k], w_v[(size_t)k * (H_ * HD_) + h * HD_ + j], acc);
    oh[(size_t)b * H_ * HD_ + h * HD_ + j] = acc;
}

// -------------------------------------------------------- output projection
__global__ void mla_oproj_kernel(const float* __restrict__ oh,
                                 const float* __restrict__ w_o,
                                 float* __restrict__ out) {
    __shared__ float oh_s[H_ * HD_];
    const int b = blockIdx.x;
    const int d = blockIdx.y * 512 + threadIdx.x * 2;
    for (int i = threadIdx.x; i < H_ * HD_; i += blockDim.x)
        oh_s[i] = oh[(size_t)b * H_ * HD_ + i];
    __syncthreads();
    float a0 = 0.0f, a1 = 0.0f;
    for (int j = 0; j < H_ * HD_; ++j) {
        const float o = oh_s[j];
        const float2 w = *(const float2*)(w_o + (size_t)j * D_ + d);
        a0 = fmaf(o, w.x, a0);
        a1 = fmaf(o, w.y, a1);
    }
    out[(size_t)b * D_ + d]     = a0;
    out[(size_t)b * D_ + d + 1] = a1;
}

// ---------------------------------------------------------------- launcher
extern "C" void kernel_launch(void* const* d_in, const int* in_sizes, int n_in,
                              void* d_out, int out_size, void* d_ws, size_t ws_size,
                              hipStream_t stream) {
    const float* x      = (const float*)d_in[0];   // [8,1,4096]
    const float* past_c = (const float*)d_in[1];   // [8,8191,512]
    const float* aqk    = (const float*)d_in[2];   // [32,4096,512]
    const float* w_c    = (const float*)d_in[3];   // [4096,512]
    const float* w_v    = (const float*)d_in[4];   // [512,4096]
    const float* w_o    = (const float*)d_in[5];   // [4096,4096]
    float* out = (float*)d_out;                    // [8,1,4096]

    float* ws       = (float*)d_ws;
    float* q_abs_ws = ws;                                        // 8*32*512
    float* c_ws     = q_abs_ws + (size_t)B_ * H_ * KV_;          // 8*512
    float* ctx_part = c_ws + (size_t)B_ * KV_;                   // 8*8*32*512
    float* ml_ws    = ctx_part + (size_t)B_ * NCH_ * H_ * KV_;   // 8*8*2*32
    float* ctx_comb = ml_ws + (size_t)B_ * NCH_ * 2 * H_;        // 8*32*512
    float* oh_ws    = ctx_comb + (size_t)B_ * H_ * KV_;          // 8*4096

    // 1) zero the atomic accumulators (q_abs + c)
    const int nzero = B_ * H_ * KV_ + B_ * KV_;
    mla_zero_kernel<<<(nzero + 255) / 256, 256, 0, stream>>>(q_abs_ws, nzero);

    // 2) q_abs = x @ absorbed_qk (streams 256MB once), c = x @ w_compress
    mla_proj8_kernel<<<dim3(H_, 4), 128, 0, stream>>>(aqk, (size_t)D_ * KV_, x,
                                                      q_abs_ws, H_ * KV_);
    mla_proj8_kernel<<<dim3(1, 4), 128, 0, stream>>>(w_c, 0, x, c_ws, KV_);

    // 3) flash decode over the compressed cache (WMMA scores + ctx)
    const size_t smem_bytes = (size_t)(H_ * KV_ + TILE_ * KV_ + H_ * TILE_ + 8 * H_) * sizeof(float);
    mla_flash_kernel<<<dim3(B_, NCH_), 128, smem_bytes, stream>>>(past_c, c_ws, q_abs_ws,
                                                                  ctx_part, ml_ws);

    // 4) combine chunk partials
    mla_combine_kernel<<<dim3(B_, H_), 128, 0, stream>>>(ctx_part, ml_ws, ctx_comb);

    // 5) per-head V up-projection, 6) output projection
    mla_vup_kernel<<<dim3(B_, H_), 128, 0, stream>>>(ctx_comb, w_v, oh_ws);
    mla_oproj_kernel<<<dim3(B_, 8), 256, 0, stream>>>(oh_ws, w_o, out);
}